// LNN_2989297238150
// MI455X (gfx1250) — compile-verified
//
#include <hip/hip_runtime.h>
#include <hip/hip_bf16.h>
#include <math.h>

// ---------------------------------------------------------------------------
// LNN derivative kernel for MI455X (gfx1250), fp32 WMMA (V_WMMA_F32_16X16X4_F32)
// Each wave handles 16 batch elements. Activations live as a 32x64 fp32 panel
// in LDS: columns [0..15] = value channel, [16..31]/[32..47]/[48..63] = JVP
// tangents d/dq0, d/dq1, d/dq2. Weights pre-packed to zero-padded 32x32 tiles.
// ---------------------------------------------------------------------------

typedef __attribute__((ext_vector_type(2))) float v2f;
typedef __attribute__((ext_vector_type(8))) float v8f;

#define WPB   2        // waves per block (64 threads)
#define NCOL  64       // activation panel columns
#define LSTRIDE 1056   // floats per packed layer: 32*32 weights + 32 bias

struct DampP { const float* W[6]; const float* b[6]; };

__device__ __forceinline__ void lds_fence() {
  // LDS ops from one wave are in-order (DScnt); this is a belt-and-braces
  // HW wait + compiler memory barrier between panel phases.
  asm volatile("s_wait_dscnt 0" ::: "memory");
}

// A fragment: 16x4 fp32. lane&15 = M, lane>>4 selects K pair {0,1}/{2,3}.
__device__ __forceinline__ v2f load_a_frag(const float* __restrict__ Wp, int m0, int k0, int lane) {
  int m = m0 + (lane & 15);
  int k = k0 + ((lane >> 4) << 1);
  return *reinterpret_cast<const v2f*>(Wp + m * 32 + k);   // 8B aligned (k0 multiple of 4)
}

// B fragment: 4x16 fp32. lane&15 = N, lane>>4 selects K-row pair.
__device__ __forceinline__ v2f load_b_frag(const float* Xl, int k0, int n0, int lane) {
  int k = k0 + ((lane >> 4) << 1);
  int n = n0 + (lane & 15);
  v2f b;
  b.x = Xl[k * NCOL + n];
  b.y = Xl[(k + 1) * NCOL + n];
  return b;
}

// D fragment: 16x16 fp32. vgpr r holds row (m0 + 8*(lane>>4) + r), col lane&15.
__device__ __forceinline__ void store_d_frag(float* Zl, int m0, int n0, int lane, v8f d) {
  int n  = n0 + (lane & 15);
  int mb = m0 + ((lane >> 4) << 3);
#pragma unroll
  for (int r = 0; r < 8; ++r) Zl[(mb + r) * NCOL + n] = d[r];
}

// One dense layer: Z(32x64) = Wpad(32x32) * X(32x64), MT m-tiles, KS k-steps of 4.
template <int MT, int KS>
__device__ __forceinline__ void gemm_layer(const float* __restrict__ Wp,
                                           const float* Xl, float* Zl, int lane) {
  v2f af[MT][KS];
#pragma unroll
  for (int mt = 0; mt < MT; ++mt)
#pragma unroll
    for (int k = 0; k < KS; ++k) af[mt][k] = load_a_frag(Wp, mt * 16, k * 4, lane);

#pragma unroll
  for (int nt = 0; nt < 4; ++nt) {
    v2f bf[KS];
#pragma unroll
    for (int k = 0; k < KS; ++k) bf[k] = load_b_frag(Xl, k * 4, nt * 16, lane);
#pragma unroll
    for (int mt = 0; mt < MT; ++mt) {
      v8f acc = {0.f, 0.f, 0.f, 0.f, 0.f, 0.f, 0.f, 0.f};
#pragma unroll
      for (int k = 0; k < KS; ++k)
        acc = __builtin_amdgcn_wmma_f32_16x16x4_f32(
            /*neg_a=*/false, af[mt][k], /*neg_b=*/false, bf[k],
            /*c_mod=*/(short)0, acc, /*reuse_a=*/false, /*reuse_b=*/false);
      store_d_frag(Zl, mt * 16, nt * 16, lane, acc);
    }
  }
}

// value cols: h = softplus(z + b); tangent cols: t = sigmoid(z_value + b) * z_t
// Each lane owns column (lane&15) and rows r0..r0+15, so sigmoid stays in regs.
__device__ __forceinline__ void elw_softplus(const float* Zl, float* Xl,
                                             const float* __restrict__ bias, int lane) {
  int n  = lane & 15;
  int r0 = (lane >> 4) * 16;
  float sg[16];
#pragma unroll
  for (int r = 0; r < 16; ++r) {
    float z = Zl[(r0 + r) * NCOL + n] + bias[r0 + r];
    sg[r] = 1.f / (1.f + expf(-z));
    Xl[(r0 + r) * NCOL + n] = fmaxf(z, 0.f) + log1pf(expf(-fabsf(z)));
  }
#pragma unroll
  for (int j = 0; j < 3; ++j) {
    int c = 16 * (1 + j) + n;
#pragma unroll
    for (int r = 0; r < 16; ++r) {
      float z = Zl[(r0 + r) * NCOL + c];
      Xl[(r0 + r) * NCOL + c] = sg[r] * z;
    }
  }
}

// Panel init: rows 0..3 (K=4, row3 zero-pad); value cols = q, tangent cols = I3.
__device__ __forceinline__ void init_panel(float* Xl, const float q[3], int lane) {
  if (lane < 16) {
    int n = lane;
    Xl[0 * NCOL + n] = q[0];
    Xl[1 * NCOL + n] = q[1];
    Xl[2 * NCOL + n] = q[2];
    Xl[3 * NCOL + n] = 0.f;
#pragma unroll
    for (int j = 0; j < 3; ++j) {
      int c = 16 * (1 + j) + n;
#pragma unroll
      for (int r = 0; r < 4; ++r) Xl[r * NCOL + c] = (r == j) ? 1.f : 0.f;
    }
  }
}

template <int MT_LAST>
__device__ __forceinline__ void mlp_fwd_jvp(const float* __restrict__ wp4,
                                            float* Xl, float* Zl, int lane) {
  gemm_layer<2, 1>(wp4 + 0 * LSTRIDE, Xl, Zl, lane); lds_fence();
  elw_softplus(Zl, Xl, wp4 + 0 * LSTRIDE + 1024, lane); lds_fence();
  gemm_layer<2, 8>(wp4 + 1 * LSTRIDE, Xl, Zl, lane); lds_fence();
  elw_softplus(Zl, Xl, wp4 + 1 * LSTRIDE + 1024, lane); lds_fence();
  gemm_layer<2, 8>(wp4 + 2 * LSTRIDE, Xl, Zl, lane); lds_fence();
  elw_softplus(Zl, Xl, wp4 + 2 * LSTRIDE + 1024, lane); lds_fence();
  gemm_layer<MT_LAST, 8>(wp4 + 3 * LSTRIDE, Xl, Zl, lane); lds_fence();
}

__global__ void __launch_bounds__(32 * WPB)
lnn_kernel(const float* __restrict__ s, const float* __restrict__ acc_in,
           const float* __restrict__ wp, DampP dp, float* __restrict__ out, int B) {
  __shared__ float Xs[WPB][32 * NCOL];
  __shared__ float Zs[WPB][32 * NCOL];

  const int wave = threadIdx.x >> 5;
  const int lane = threadIdx.x & 31;
  float* X = Xs[wave];
  float* Z = Zs[wave];

  const int e0 = ((int)blockIdx.x * WPB + wave) * 16;
  int e = e0 + (lane & 15);
  if (e >= B) e = B - 1;  // clamp (keeps EXEC full; duplicate writes are identical)

  __builtin_prefetch(wp, 0, 3);

  float q[3], qd[3], av[3];
#pragma unroll
  for (int i = 0; i < 3; ++i) {
    q[i]  = s[e * 6 + i];
    qd[i] = s[e * 6 + 3 + i];
    av[i] = acc_in[e * 3 + i];
  }

  const float* wpm = wp;                // mass: 4 packed layers
  const float* wpp = wp + 4 * LSTRIDE;  // pot : 4 packed layers

  // ---------------- mass MLP + Jacobian (L, G_j) ----------------
  init_panel(X, q, lane); lds_fence();
  mlp_fwd_jvp<1>(wpm, X, Z, lane);

  float yv[6], gy[3][6];
  {
    int n = lane & 15;
    const float* b4 = wpm + 3 * LSTRIDE + 1024;
#pragma unroll
    for (int t = 0; t < 6; ++t) {
      yv[t] = Z[t * NCOL + n] + b4[t];
#pragma unroll
      for (int j = 0; j < 3; ++j) gy[j][t] = Z[t * NCOL + 16 * (1 + j) + n];
    }
  }

  // ---------------- potential MLP gradient ----------------
  init_panel(X, q, lane); lds_fence();
  mlp_fwd_jvp<1>(wpp, X, Z, lane);

  float dV[3];
  {
    int n = lane & 15;
#pragma unroll
    for (int j = 0; j < 3; ++j) dV[j] = Z[0 * NCOL + 16 * (1 + j) + n];
  }

  // ---------------- damping MLP (width 5, tanh) — scalar VALU ----------------
  float h[5], hn[5];
#pragma unroll
  for (int o = 0; o < 5; ++o) {
    float z = dp.b[0][o];
#pragma unroll
    for (int i = 0; i < 3; ++i) z += dp.W[0][o * 3 + i] * q[i];
    h[o] = tanhf(z);
  }
#pragma unroll
  for (int l = 1; l < 5; ++l) {
#pragma unroll
    for (int o = 0; o < 5; ++o) {
      float z = dp.b[l][o];
#pragma unroll
      for (int i = 0; i < 5; ++i) z += dp.W[l][o * 5 + i] * h[i];
      hn[o] = tanhf(z);
    }
#pragma unroll
    for (int o = 0; o < 5; ++o) h[o] = hn[o];
  }
  float yd[6];
#pragma unroll
  for (int o = 0; o < 6; ++o) {
    float z = dp.b[5][o];
#pragma unroll
    for (int i = 0; i < 5; ++i) z += dp.W[5][o * 5 + i] * h[i];
    yd[o] = z;
  }

  // ---------------- 3x3 tensor algebra ----------------
  // tril order: (0,0)(1,0)(1,1)(2,0)(2,1)(2,2)
  float Lm[3][3] = {{yv[0], 0.f, 0.f}, {yv[1], yv[2], 0.f}, {yv[3], yv[4], yv[5]}};
  float Gm[3][3][3];
#pragma unroll
  for (int j = 0; j < 3; ++j) {
    Gm[j][0][0] = gy[j][0]; Gm[j][0][1] = 0.f;      Gm[j][0][2] = 0.f;
    Gm[j][1][0] = gy[j][1]; Gm[j][1][1] = gy[j][2]; Gm[j][1][2] = 0.f;
    Gm[j][2][0] = gy[j][3]; Gm[j][2][1] = gy[j][4]; Gm[j][2][2] = gy[j][5];
  }

  // dM[i][a][b] = d(L L^T)[a][b] / dq_i = sum_k G_i[a][k] L[b][k] + L[a][k] G_i[b][k]
  float dM[3][3][3];
#pragma unroll
  for (int i = 0; i < 3; ++i)
#pragma unroll
    for (int a = 0; a < 3; ++a)
#pragma unroll
      for (int b = 0; b < 3; ++b) {
        float v = 0.f;
#pragma unroll
        for (int k = 0; k < 3; ++k)
          v += Gm[i][a][k] * Lm[b][k] + Lm[a][k] * Gm[i][b][k];
        dM[i][a][b] = v;
      }

  // c[i] = sum_{j,k} dM[j][i][k] qd_k qd_j - 0.5 * sum_{k,b} dM[i][k][b] qd_k qd_b
  float c[3];
#pragma unroll
  for (int i = 0; i < 3; ++i) {
    float c1 = 0.f, c2 = 0.f;
#pragma unroll
    for (int j = 0; j < 3; ++j)
#pragma unroll
      for (int k = 0; k < 3; ++k) {
        c1 += dM[j][i][k] * qd[k] * qd[j];
        c2 += dM[i][j][k] * qd[j] * qd[k];
      }
    c[i] = c1 - 0.5f * c2;
  }

  // D = 0.4 * tril(yd) tril(yd)^T ; damping = D qd
  float Yd[3][3] = {{yd[0], 0.f, 0.f}, {yd[1], yd[2], 0.f}, {yd[3], yd[4], yd[5]}};
  float damping[3];
#pragma unroll
  for (int i = 0; i < 3; ++i) {
    float v = 0.f;
#pragma unroll
    for (int j = 0; j < 3; ++j) {
      float Dij = 0.f;
#pragma unroll
      for (int k = 0; k < 3; ++k) Dij += Yd[i][k] * Yd[j][k];
      v += 0.4f * Dij * qd[j];
    }
    damping[i] = v;
  }

  float rhs[3];
#pragma unroll
  for (int i = 0; i < 3; ++i) rhs[i] = av[i] - c[i] - dV[i] - damping[i];

  // cho_solve with lower L: L z = rhs ; L^T x = z
  float z0 = rhs[0] / Lm[0][0];
  float z1 = (rhs[1] - Lm[1][0] * z0) / Lm[1][1];
  float z2 = (rhs[2] - Lm[2][0] * z0 - Lm[2][1] * z1) / Lm[2][2];
  float x2 = z2 / Lm[2][2];
  float x1 = (z1 - Lm[2][1] * x2) / Lm[1][1];
  float x0 = (z0 - Lm[1][0] * x1 - Lm[2][0] * x2) / Lm[0][0];

  if (lane < 16) {
    out[e * 6 + 0] = qd[0];
    out[e * 6 + 1] = qd[1];
    out[e * 6 + 2] = qd[2];
    out[e * 6 + 3] = x0;
    out[e * 6 + 4] = x1;
    out[e * 6 + 5] = x2;
  }
}

// Pack a (rows x cols) row-major weight + bias into zero-padded 32x32 + 32.
__global__ void pack_w(const float* __restrict__ W, const float* __restrict__ b,
                       float* __restrict__ dst, int rows, int cols) {
  for (int t = threadIdx.x; t < LSTRIDE; t += blockDim.x) {
    if (t < 1024) {
      int r = t >> 5, cc = t & 31;
      dst[t] = (r < rows && cc < cols) ? W[r * cols + cc] : 0.f;
    } else {
      int r = t - 1024;
      dst[t] = (r < rows) ? b[r] : 0.f;
    }
  }
}

extern "C" void kernel_launch(void* const* d_in, const int* in_sizes, int n_in,
                              void* d_out, int out_size, void* d_ws, size_t ws_size,
                              hipStream_t stream) {
  const float* s = (const float*)d_in[0];
  const float* a = (const float*)d_in[1];

  const float* mass[8] = {};
  const float* pot[8]  = {};
  const float* damp[12] = {};

  if (n_in >= 30) {
    // Classify flattened param leaves by element count (robust to pytree order).
    int i = 2;
    while (i < n_in) {
      int sz = in_sizes[i];
      if (sz == 15) {                       // damp starts with W(5x3)=15
        for (int k = 0; k < 12 && i + k < n_in; ++k) damp[k] = (const float*)d_in[i + k];
        i += 12;
      } else if (sz == 90) {                // mass or pot: both start with 90
        bool is_mass = (i + 6 < n_in) && (in_sizes[i + 6] == 180);
        if (is_mass) { for (int k = 0; k < 8; ++k) mass[k] = (const float*)d_in[i + k]; }
        else         { for (int k = 0; k < 8; ++k) pot[k]  = (const float*)d_in[i + k]; }
        i += 8;
      } else {
        ++i;
      }
    }
  } else {
    // Fallback: single concatenated blob in dict insertion order (mass,pot,damp).
    const float* p = (const float*)d_in[2];
    static const int msz[8]  = {90, 30, 900, 30, 900, 30, 180, 6};
    static const int psz[8]  = {90, 30, 900, 30, 900, 30, 30, 1};
    static const int dsz[12] = {15, 5, 25, 5, 25, 5, 25, 5, 25, 5, 30, 6};
    for (int k = 0; k < 8; ++k)  { mass[k] = p; p += msz[k]; }
    for (int k = 0; k < 8; ++k)  { pot[k]  = p; p += psz[k]; }
    for (int k = 0; k < 12; ++k) { damp[k] = p; p += dsz[k]; }
  }

  float* wp = (float*)d_ws;  // 8 * 1056 floats = 33 KB of scratch
  static const int rows[8] = {30, 30, 30, 6, 30, 30, 30, 1};
  static const int cols[8] = {3, 30, 30, 30, 3, 30, 30, 30};
  for (int l = 0; l < 8; ++l) {
    const float* W = (l < 4) ? mass[2 * l]     : pot[2 * (l - 4)];
    const float* b = (l < 4) ? mass[2 * l + 1] : pot[2 * (l - 4) + 1];
    pack_w<<<dim3(1), dim3(256), 0, stream>>>(W, b, wp + l * LSTRIDE, rows[l], cols[l]);
  }

  DampP dp;
  for (int k = 0; k < 6; ++k) { dp.W[k] = damp[2 * k]; dp.b[k] = damp[2 * k + 1]; }

  int B = in_sizes[0] / 6;
  int nblk = (B + 16 * WPB - 1) / (16 * WPB);
  lnn_kernel<<<dim3(nblk), dim3(32 * WPB), 0, stream>>>(s, a, (const float*)wp, dp,
                                                        (float*)d_out, B);
}